// Net_80058190398073
// MI455X (gfx1250) — compile-verified
//
#include <hip/hip_runtime.h>
#include <hip/hip_bf16.h>

typedef float v8f __attribute__((ext_vector_type(8)));
typedef float v2f __attribute__((ext_vector_type(2)));

#define N_NODES 100000
#define N_EDGES 3200000
#define N_GRAPHS 512

// ---------------------------------------------------------------------------
// Small utility kernels
// ---------------------------------------------------------------------------
__global__ void k_fill(float* __restrict__ p, float v, int n) {
    int i = blockIdx.x * blockDim.x + threadIdx.x;
    if (i < n) p[i] = v;
}

__global__ void k_deg(const int* __restrict__ dst, float* __restrict__ deg, int e) {
    int i = blockIdx.x * blockDim.x + threadIdx.x;
    if (i < e) {
        __hip_atomic_fetch_add(&deg[dst[i]], 1.0f,
                               __ATOMIC_RELAXED, __HIP_MEMORY_SCOPE_AGENT);
    }
}

__global__ void k_rsqrt(float* __restrict__ d, int n) {
    int i = blockIdx.x * blockDim.x + threadIdx.x;
    if (i < n) d[i] = rsqrtf(d[i]);
}

// ---------------------------------------------------------------------------
// Node-feature transform: out[n][COUT] = in[n][CIN] @ W[CIN][COUT]
// One wave per 16-node tile, FP32 WMMA 16x16x4. K padded with zeros, N tiled
// by 16 (COUT in {16,32,48} -> exact tiles). nrows % 16 == 0 (100000 = 6250*16).
// A layout (ISA 7.12.2): lanes 0-15 -> K=kk,kk+1 ; lanes 16-31 -> K=kk+2,kk+3.
// D layout: VGPR i -> M=i (lanes 0-15), M=i+8 (lanes 16-31), N = lane&15.
// ---------------------------------------------------------------------------
template <int CIN, int COUT>
__global__ void k_transform(const float* __restrict__ in,
                            const float* __restrict__ W,
                            float* __restrict__ out, int nrows) {
    int wave = (blockIdx.x * blockDim.x + threadIdx.x) >> 5;
    int lane = threadIdx.x & 31;
    int row0 = wave * 16;
    if (row0 >= nrows) return;          // wave-uniform: EXEC all-ones for WMMA
    const int half = lane >> 4;         // 0: lanes 0-15, 1: lanes 16-31
    const int l16  = lane & 15;
    const int r    = row0 + l16;

    for (int nt = 0; nt < COUT; nt += 16) {
        v8f c = {};
        const int col = nt + l16;
#pragma unroll
        for (int kk = 0; kk < CIN; kk += 4) {
            const int ka = kk + half * 2;
            v2f a, b;
            a.x = (ka     < CIN) ? in[r * CIN + ka]     : 0.0f;
            a.y = (ka + 1 < CIN) ? in[r * CIN + ka + 1] : 0.0f;
            b.x = (ka     < CIN) ? W[ka       * COUT + col] : 0.0f;
            b.y = (ka + 1 < CIN) ? W[(ka + 1) * COUT + col] : 0.0f;
            c = __builtin_amdgcn_wmma_f32_16x16x4_f32(
                    false, a, false, b, (short)0, c, false, false);
        }
#pragma unroll
        for (int i = 0; i < 8; ++i) {
            out[(size_t)(row0 + i + half * 8) * COUT + col] = c[i];
        }
    }
}

// ---------------------------------------------------------------------------
// Edge scatter: acc[dst] += dis[src]*dis[dst] * t[src], float4-vectorized.
// One thread per (edge, 4-channel group); consecutive threads share an edge
// so the gathered float4s and the atomic targets are contiguous.
// ---------------------------------------------------------------------------
template <int C>
__global__ void k_scatter(const int* __restrict__ src, const int* __restrict__ dst,
                          const float* __restrict__ dis,
                          const float* __restrict__ t,
                          float* __restrict__ acc, int e) {
    constexpr int G = C / 4;
    int gid = blockIdx.x * blockDim.x + threadIdx.x;
    if (gid >= e * G) return;
    const int ei = gid / G;
    const int g  = gid - ei * G;
    const int s  = src[ei];
    const int d  = dst[ei];
    const float nrm = dis[s] * dis[d];
    const float4 v = *(const float4*)(t + (size_t)s * C + g * 4);
    float* p = acc + (size_t)d * C + g * 4;
    __hip_atomic_fetch_add(p + 0, nrm * v.x, __ATOMIC_RELAXED, __HIP_MEMORY_SCOPE_AGENT);
    __hip_atomic_fetch_add(p + 1, nrm * v.y, __ATOMIC_RELAXED, __HIP_MEMORY_SCOPE_AGENT);
    __hip_atomic_fetch_add(p + 2, nrm * v.z, __ATOMIC_RELAXED, __HIP_MEMORY_SCOPE_AGENT);
    __hip_atomic_fetch_add(p + 3, nrm * v.w, __ATOMIC_RELAXED, __HIP_MEMORY_SCOPE_AGENT);
}

// ---------------------------------------------------------------------------
// Finalize: out = relu(acc + dis^2 * t + b)   (self-loop term folded in here)
// ---------------------------------------------------------------------------
template <int C>
__global__ void k_finalize(const float* __restrict__ acc, const float* __restrict__ t,
                           const float* __restrict__ dis, const float* __restrict__ b,
                           float* __restrict__ out, int n) {
    int idx = blockIdx.x * blockDim.x + threadIdx.x;
    if (idx >= n * C) return;
    const int i = idx / C;
    const int c = idx - i * C;
    const float di = dis[i];
    const float v  = acc[idx] + di * di * t[idx] + b[c];
    out[idx] = fmaxf(v, 0.0f);
}

// ---------------------------------------------------------------------------
// Segment-max pool. Inputs are post-ReLU (>= 0), pool init'd to 0, so integer
// atomicMax on the float bit pattern is order-correct and also implements the
// "empty segment -> 0" rule.
// ---------------------------------------------------------------------------
__global__ void k_pool(const float* __restrict__ h, const int* __restrict__ batch,
                       float* __restrict__ pool, int n) {
    int idx = blockIdx.x * blockDim.x + threadIdx.x;
    if (idx >= n * 48) return;
    const int i = idx / 48;
    const int c = idx - i * 48;
    const int g = batch[i];
    atomicMax((int*)&pool[g * 48 + c], __float_as_int(h[idx]));
}

// ---------------------------------------------------------------------------
// Final MLP: out[512][10] = relu(pool @ Wl1 + bl1) @ Wl2 + bl2
// ---------------------------------------------------------------------------
__global__ void k_mlp(const float* __restrict__ pool,
                      const float* __restrict__ Wl1, const float* __restrict__ bl1,
                      const float* __restrict__ Wl2, const float* __restrict__ bl2,
                      float* __restrict__ out) {
    int g = blockIdx.x * blockDim.x + threadIdx.x;
    if (g >= N_GRAPHS) return;
    float p[48];
#pragma unroll
    for (int k = 0; k < 48; ++k) p[k] = pool[g * 48 + k];
    float hid[24];
#pragma unroll
    for (int j = 0; j < 24; ++j) {
        float s = bl1[j];
#pragma unroll
        for (int k = 0; k < 48; ++k) s += p[k] * Wl1[k * 24 + j];
        hid[j] = fmaxf(s, 0.0f);
    }
#pragma unroll
    for (int o = 0; o < 10; ++o) {
        float s = bl2[o];
#pragma unroll
        for (int j = 0; j < 24; ++j) s += hid[j] * Wl2[j * 10 + o];
        out[g * 10 + o] = s;
    }
}

// ---------------------------------------------------------------------------
static inline int cdiv(int a, int b) { return (a + b - 1) / b; }

extern "C" void kernel_launch(void* const* d_in, const int* in_sizes, int n_in,
                              void* d_out, int out_size, void* d_ws, size_t ws_size,
                              hipStream_t stream) {
    const float* x     = (const float*)d_in[0];
    const int*   ei    = (const int*)d_in[1];
    const int*   batch = (const int*)d_in[2];
    const float* W1  = (const float*)d_in[3];
    const float* b1  = (const float*)d_in[4];
    const float* W2  = (const float*)d_in[5];
    const float* b2  = (const float*)d_in[6];
    const float* W3  = (const float*)d_in[7];
    const float* b3  = (const float*)d_in[8];
    const float* Wl1 = (const float*)d_in[9];
    const float* bl1 = (const float*)d_in[10];
    const float* Wl2 = (const float*)d_in[11];
    const float* bl2 = (const float*)d_in[12];

    const int E = in_sizes[1] / 2;
    const int N = in_sizes[0] / 2;
    const int* src = ei;        // edge_index[0]
    const int* dst = ei + E;    // edge_index[1]

    // Workspace layout (floats): dis | F(N*48) | T(N*48) | ACC(N*48) | POOL
    float* ws   = (float*)d_ws;
    float* DIS  = ws;
    float* F    = ws + 100352;
    float* T    = F  + (size_t)N_NODES * 48;
    float* ACC  = T  + (size_t)N_NODES * 48;
    float* POOL = ACC + (size_t)N_NODES * 48;

    const int BLK = 256;

    // Degree + symmetric normalization (self-loop contributes the initial 1.0)
    k_fill<<<cdiv(N, BLK), BLK, 0, stream>>>(DIS, 1.0f, N);
    k_deg<<<cdiv(E, BLK), BLK, 0, stream>>>(dst, DIS, E);
    k_rsqrt<<<cdiv(N, BLK), BLK, 0, stream>>>(DIS, N);

    const int tblocks = cdiv((N / 16) * 32, BLK);   // one wave per 16-node tile

    // ---- Layer 1: 2 -> 16 ----
    k_transform<2, 16><<<tblocks, BLK, 0, stream>>>(x, W1, F, N);
    hipMemsetAsync(ACC, 0, (size_t)N * 16 * sizeof(float), stream);
    k_scatter<16><<<cdiv(E * 4, BLK), BLK, 0, stream>>>(src, dst, DIS, F, ACC, E);
    k_finalize<16><<<cdiv(N * 16, BLK), BLK, 0, stream>>>(ACC, F, DIS, b1, F, N);

    // ---- Layer 2: 16 -> 32 ----
    k_transform<16, 32><<<tblocks, BLK, 0, stream>>>(F, W2, T, N);
    hipMemsetAsync(ACC, 0, (size_t)N * 32 * sizeof(float), stream);
    k_scatter<32><<<cdiv(E * 8, BLK), BLK, 0, stream>>>(src, dst, DIS, T, ACC, E);
    k_finalize<32><<<cdiv(N * 32, BLK), BLK, 0, stream>>>(ACC, T, DIS, b2, T, N);

    // ---- Layer 3: 32 -> 48 ----
    k_transform<32, 48><<<tblocks, BLK, 0, stream>>>(T, W3, F, N);
    hipMemsetAsync(ACC, 0, (size_t)N * 48 * sizeof(float), stream);
    k_scatter<48><<<cdiv(E * 12, BLK), BLK, 0, stream>>>(src, dst, DIS, F, ACC, E);
    k_finalize<48><<<cdiv(N * 48, BLK), BLK, 0, stream>>>(ACC, F, DIS, b3, F, N);

    // ---- Pool + MLP head ----
    hipMemsetAsync(POOL, 0, (size_t)N_GRAPHS * 48 * sizeof(float), stream);
    k_pool<<<cdiv(N * 48, BLK), BLK, 0, stream>>>(F, batch, POOL, N);
    k_mlp<<<cdiv(N_GRAPHS, BLK), BLK, 0, stream>>>(POOL, Wl1, bl1, Wl2, bl2,
                                                   (float*)d_out);
}